// TemporalBuffer_79362405695873
// MI455X (gfx1250) — compile-verified
//
#include <hip/hip_runtime.h>

// Event-histogram scatter for MI455X (gfx1250, wave32).
// Bandwidth path: B128 non-temporal streaming loads of the 5 event arrays,
// L2-resident float atomics into the 3.46MB active histogram (192MB L2),
// global_prefetch_b8 to cover the grid-stride loop.

typedef int   v4i __attribute__((ext_vector_type(4)));
typedef float v4f __attribute__((ext_vector_type(4)));

#define N_EVENT_STEPS 10
#define N_PROP_STEPS  10
#define HEIGHT        180
#define WIDTH         240

// ---------------------------------------------------------------------------
// PROBE (never launched): first function in file so the disasm snippet shows
// exactly how the toolchain lowers (a) a non-temporal float load and (b) a
// discarded-result float atomicAdd on gfx1250. Expect:
//     global_load_b32 ... th:TH_LOAD_NT
//     global_atomic_add_f32 ...            (native, no CAS loop, no return)
// ---------------------------------------------------------------------------
__global__ void tb_probe_kernel(float* __restrict__ out,
                                const float* __restrict__ v) {
    float val = __builtin_nontemporal_load(v + threadIdx.x);
    (void)atomicAdd(&out[threadIdx.x], val);
}

// ---------------------------------------------------------------------------
// Scatter-add events into the histogram.
// ---------------------------------------------------------------------------
__device__ __forceinline__ void tb_accum_one(int xv, int yv, int pv, int tv,
                                             float val, bool active, bool remap,
                                             float tminf, float denom,
                                             float steps,
                                             float* __restrict__ out) {
    float tf = (float)tv;
    float tn = active ? ((tf - tminf) / denom) * steps : 0.0f;
    int ti = (int)tn;                       // trunc toward zero (astype int32)
    ti = min(max(ti, 0), N_EVENT_STEPS - 1);
    int xi = min(max(xv, 0), WIDTH - 1);
    int yi = min(max(yv, 0), HEIGHT - 1);
    int c  = remap ? ((1 - pv) >> 1) : pv;  // >>1 == floor-div-2 (matches //)
    c = min(max(c, 0), 1);
    int idx = ((ti * 2 + c) * HEIGHT + yi) * WIDTH + xi;
    (void)atomicAdd(&out[idx], val);
}

__global__ void tb_scatter_kernel(const int* __restrict__ x,
                                  const int* __restrict__ y,
                                  const int* __restrict__ p,
                                  const int* __restrict__ t,
                                  const float* __restrict__ vals,
                                  float* __restrict__ out,
                                  const int* __restrict__ ws, int n) {
    // Uniform normalization constants. t is sorted by construction, so
    // min == t[0], max == t[n-1] (SMEM scalar loads, no reduction pass).
    const float STEPS = (float)(10.0 - 1e-6);   // matches N_EVENT_STEPS - 1e-6
    float tminf = (float)t[0];
    float tmaxf = (float)t[n - 1];
    float denom = fmaxf(tmaxf - tminf, 1e-9f);
    bool  active = tmaxf > tminf;
    bool  remap  = ws[0] < 0;

    int stride = gridDim.x * blockDim.x;
    int gid    = blockIdx.x * blockDim.x + threadIdx.x;
    int n4     = n >> 2;

    for (int vi = gid; vi < n4; vi += stride) {
        // Prefetch next grid-stride tile (global_prefetch_b8).
        if (vi + stride < n4) {
            __builtin_prefetch((const v4i*)t + vi + stride, 0, 1);
            __builtin_prefetch((const v4i*)x + vi + stride, 0, 1);
            __builtin_prefetch((const v4i*)y + vi + stride, 0, 1);
            __builtin_prefetch((const v4i*)p + vi + stride, 0, 1);
            __builtin_prefetch((const v4f*)vals + vi + stride, 0, 1);
        }
        // Streaming B128 non-temporal loads: keep L2 for the histogram.
        v4i xv = __builtin_nontemporal_load((const v4i*)x + vi);
        v4i yv = __builtin_nontemporal_load((const v4i*)y + vi);
        v4i pv = __builtin_nontemporal_load((const v4i*)p + vi);
        v4i tv = __builtin_nontemporal_load((const v4i*)t + vi);
        v4f vv = __builtin_nontemporal_load((const v4f*)vals + vi);
#pragma unroll
        for (int k = 0; k < 4; ++k)
            tb_accum_one(xv[k], yv[k], pv[k], tv[k], vv[k],
                         active, remap, tminf, denom, STEPS, out);
    }
    // Scalar tail (n is a multiple of 4 in practice; kept for generality).
    for (int i = (n4 << 2) + gid; i < n; i += stride)
        tb_accum_one(x[i], y[i], p[i], t[i], vals[i],
                     active, remap, tminf, denom, STEPS, out);
}

// ---------------------------------------------------------------------------
// Global min over p (for the reference's p.min()<0 remap branch).
// Wave32 shuffle reduction + one atomicMin per wave.
// ---------------------------------------------------------------------------
__global__ void tb_pmin_kernel(const int* __restrict__ p, int n,
                               int* __restrict__ ws) {
    int stride = gridDim.x * blockDim.x;
    int gid    = blockIdx.x * blockDim.x + threadIdx.x;
    int lmin   = 0x7FFFFFFF;
    int n4     = n >> 2;
    for (int vi = gid; vi < n4; vi += stride) {
        v4i v = __builtin_nontemporal_load((const v4i*)p + vi);
        lmin  = min(lmin, min(min(v.x, v.y), min(v.z, v.w)));
    }
    for (int i = (n4 << 2) + gid; i < n; i += stride)
        lmin = min(lmin, p[i]);
    // wave32 reduction
    for (int off = 16; off > 0; off >>= 1)
        lmin = min(lmin, __shfl_down(lmin, off, 32));
    if ((threadIdx.x & 31) == 0)
        atomicMin(ws, lmin);
}

// ---------------------------------------------------------------------------
// Zero the output histogram, seed workspace (pmin accumulator).
// ---------------------------------------------------------------------------
__global__ void tb_zero_kernel(float* __restrict__ out, int n,
                               int* __restrict__ ws) {
    int i  = blockIdx.x * blockDim.x + threadIdx.x;
    int n4 = n >> 2;
    if (i < n4) {
        v4f z = {0.0f, 0.0f, 0.0f, 0.0f};
        ((v4f*)out)[i] = z;
    }
    int rem = n - (n4 << 2);
    if (i < rem) out[(n4 << 2) + i] = 0.0f;
    if (i == 0) ws[0] = 0x7FFFFFFF;
}

// ---------------------------------------------------------------------------
// Host-side launcher (graph-capture safe: only kernel launches on `stream`).
// ---------------------------------------------------------------------------
extern "C" void kernel_launch(void* const* d_in, const int* in_sizes, int n_in,
                              void* d_out, int out_size, void* d_ws, size_t ws_size,
                              hipStream_t stream) {
    const int*   x    = (const int*)d_in[0];
    const int*   y    = (const int*)d_in[1];
    const int*   p    = (const int*)d_in[2];
    const int*   t    = (const int*)d_in[3];   // int32 on device (JAX x64 off)
    const float* vals = (const float*)d_in[4];
    float*       out  = (float*)d_out;
    int*         ws   = (int*)d_ws;
    int          n    = in_sizes[0];

    // 1) zero histogram + init workspace
    int n4o     = out_size >> 2;
    int zneeded = n4o > 0 ? n4o : (out_size > 0 ? out_size : 1);
    int zblocks = (zneeded + 255) / 256;
    tb_zero_kernel<<<zblocks, 256, 0, stream>>>(out, out_size, ws);

    // 2) p.min() reduction (one streaming pass, ~3us at 23.3 TB/s)
    tb_pmin_kernel<<<1024, 256, 0, stream>>>(p, n, ws);

    // 3) scatter: 1,048,576 threads x 4 iters x 4 events (B128 loads)
    tb_scatter_kernel<<<4096, 256, 0, stream>>>(x, y, p, t, vals, out, ws, n);
}